// GraphSAGE_1829656068114
// MI455X (gfx1250) — compile-verified
//
#include <hip/hip_runtime.h>

#define N_NODES 50000
#define N_EDGES 600000
#define D_FEAT  128   // D_IN == D_HID == 128
#define D_OUT_C 64

typedef __attribute__((ext_vector_type(2))) float v2f;
typedef __attribute__((ext_vector_type(8))) float v8f;

// ---------------------------------------------------------------------------
// Zero a float region (grid-stride).
// ---------------------------------------------------------------------------
__global__ void sage_zero(float* __restrict__ p, long long n) {
    long long i = (long long)blockIdx.x * blockDim.x + threadIdx.x;
    long long stride = (long long)gridDim.x * blockDim.x;
    for (; i < n; i += stride) p[i] = 0.0f;
}

// ---------------------------------------------------------------------------
// Edge scatter: one wave (32 lanes) per edge. Each lane moves 4 contiguous
// floats (128 = 32*4) from feat[src] into sum[dst] via float atomics.
// Lane 0 optionally accumulates the destination degree count.
// ---------------------------------------------------------------------------
__global__ void sage_scatter(const float* __restrict__ feat,
                             const long long* __restrict__ src,
                             const long long* __restrict__ dst,
                             float* __restrict__ sum,
                             float* __restrict__ cnt,
                             int count_flag) {
    long long t = (long long)blockIdx.x * blockDim.x + threadIdx.x;
    int e    = (int)(t >> 5);
    int lane = (int)(t & 31);
    if (e >= N_EDGES) return;
    int s = (int)src[e];
    int d = (int)dst[e];
    if (count_flag && lane == 0) atomicAdd(cnt + d, 1.0f);
    const float4* row = (const float4*)(feat + (long long)s * D_FEAT);
    float4 v = row[lane];                       // 16B coalesced gather
    float* o = sum + (long long)d * D_FEAT + lane * 4;
    atomicAdd(o + 0, v.x);
    atomicAdd(o + 1, v.y);
    atomicAdd(o + 2, v.z);
    atomicAdd(o + 3, v.w);
}

// ---------------------------------------------------------------------------
// Fused SAGEConv GEMM epilogue:
//   out[16 rows x NCOLS] = (sum/max(cnt,1)) @ Wl^T + self @ Wr^T + bias
// One block = one 16-row node tile; one wave per 16-col N-tile.
// K = 128 consumed as 32 steps of V_WMMA_F32_16X16X4_F32 per matrix.
//
// Fragment layouts (CDNA5 ISA 7.12.2, wave32):
//   A 16x4 f32 : lane m=lane&15, VGPR{0,1} = K = (lane>>4)*2 + {0,1}
//   B 4x16 f32 : lane n=lane&15, VGPR{0,1} = K = (lane>>4)*2 + {0,1}
//   C/D 16x16  : lane n=lane&15, VGPR v -> m = v + 8*(lane>>4)
// ---------------------------------------------------------------------------
template <int NCOLS, bool RELU>
__global__ void sage_gemm(const float* __restrict__ sum,   // [N_NODES,128]
                          const float* __restrict__ cnt,   // [N_NODES]
                          const float* __restrict__ self,  // [N_NODES,128]
                          const float* __restrict__ Wl,    // [NCOLS,128]
                          const float* __restrict__ Wr,    // [NCOLS,128]
                          const float* __restrict__ bias,  // [NCOLS]
                          float* __restrict__ out)         // [N_NODES,NCOLS]
{
    __shared__ float As[16 * D_FEAT];   // mean-aggregated tile
    __shared__ float Xs[16 * D_FEAT];   // self-feature tile
    __shared__ float inv[16];

    const int row0 = blockIdx.x * 16;
    const int tid  = threadIdx.x;
    const int nthr = NCOLS * 2;         // 32 * (NCOLS/16)

    if (tid < 16) inv[tid] = 1.0f / fmaxf(cnt[row0 + tid], 1.0f);
    __syncthreads();

    // Coalesced tile staging: each thread loads strided linear indices.
    for (int i = tid; i < 16 * D_FEAT; i += nthr) {
        int r = i >> 7;                 // row within tile
        int c = i & 127;                // feature index
        long long g = (long long)(row0 + r) * D_FEAT + c;
        As[i] = sum[g] * inv[r];        // mean = sum / max(cnt,1)
        Xs[i] = self[g];
    }
    __syncthreads();

    const int lane  = tid & 31;
    const int ln15  = lane & 15;            // A-row / B-col / D-col
    const int khalf = (lane >> 4) * 2;      // K sub-pair within a 4-step
    const int n     = (tid >> 5) * 16 + ln15;

    const float* wl = Wl + (long long)n * D_FEAT;   // row n of Wl == col n of Wl^T
    const float* wr = Wr + (long long)n * D_FEAT;

    v8f acc = {};
#pragma unroll
    for (int ks = 0; ks < D_FEAT / 4; ++ks) {
        const int k = ks * 4 + khalf;   // even -> 8B aligned v2f loads
        v2f aA = *(const v2f*)&As[ln15 * D_FEAT + k];
        v2f bL = *(const v2f*)(wl + k);
        acc = __builtin_amdgcn_wmma_f32_16x16x4_f32(
            false, aA, false, bL, (short)0, acc, false, false);
        v2f aX = *(const v2f*)&Xs[ln15 * D_FEAT + k];
        v2f bR = *(const v2f*)(wr + k);
        acc = __builtin_amdgcn_wmma_f32_16x16x4_f32(
            false, aX, false, bR, (short)0, acc, false, false);
    }

    const float bv = bias[n];
#pragma unroll
    for (int v = 0; v < 8; ++v) {
        int m = v + ((lane >> 4) << 3);
        float val = acc[v] + bv;
        if (RELU) val = fmaxf(val, 0.0f);
        out[(long long)(row0 + m) * NCOLS + n] = val;
    }
}

// ---------------------------------------------------------------------------
// Host-side orchestration.
// ---------------------------------------------------------------------------
extern "C" void kernel_launch(void* const* d_in, const int* in_sizes, int n_in,
                              void* d_out, int out_size, void* d_ws, size_t ws_size,
                              hipStream_t stream) {
    const float*     x   = (const float*)d_in[0];
    const long long* ei  = (const long long*)d_in[1];   // int64 [2, E]
    const float*     W1l = (const float*)d_in[2];
    const float*     b1  = (const float*)d_in[3];
    const float*     W1r = (const float*)d_in[4];
    const float*     W2l = (const float*)d_in[5];
    const float*     b2  = (const float*)d_in[6];
    const float*     W2r = (const float*)d_in[7];
    float* out = (float*)d_out;

    const long long* src = ei;              // edge_index[0]
    const long long* dst = ei + N_EDGES;    // edge_index[1]

    // Workspace layout (bytes):
    //   sum1 [50000*128 f32]  @ 0
    //   sum2 [50000*128 f32]  @ 25.6 MB   (contiguous with sum1)
    //   h    [50000*128 f32]  @ 51.2 MB
    //   cnt  [50000 f32]      @ 76.8 MB
    const size_t FEAT_BYTES = (size_t)N_NODES * D_FEAT * sizeof(float);
    char* ws = (char*)d_ws;
    float* sum1 = (float*)(ws);
    float* sum2 = (float*)(ws + FEAT_BYTES);
    float* h    = (float*)(ws + 2 * FEAT_BYTES);
    float* cnt  = (float*)(ws + 3 * FEAT_BYTES);

    // 1) zero sum1+sum2 (contiguous) and cnt
    {
        long long nz = 2LL * N_NODES * D_FEAT;
        sage_zero<<<2048, 256, 0, stream>>>(sum1, nz);
        sage_zero<<<256, 256, 0, stream>>>(cnt, (long long)N_NODES);
    }

    // 2) layer-1 scatter (sum of neighbor features + degree count)
    {
        long long nthreads = (long long)N_EDGES * 32;
        int blocks = (int)((nthreads + 255) / 256);
        sage_scatter<<<blocks, 256, 0, stream>>>(x, src, dst, sum1, cnt, 1);
    }

    // 3) layer-1 fused GEMM + bias + ReLU -> h  (8 waves/block, N=128)
    sage_gemm<128, true><<<N_NODES / 16, 256, 0, stream>>>(
        sum1, cnt, x, W1l, W1r, b1, h);

    // 4) layer-2 scatter of h
    {
        long long nthreads = (long long)N_EDGES * 32;
        int blocks = (int)((nthreads + 255) / 256);
        sage_scatter<<<blocks, 256, 0, stream>>>(h, src, dst, sum2, cnt, 0);
    }

    // 5) layer-2 fused GEMM + bias -> out  (4 waves/block, N=64)
    sage_gemm<64, false><<<N_NODES / 16, 128, 0, stream>>>(
        sum2, cnt, h, W2l, W2r, b2, out);
}